// gat_layer_68564857914181
// MI455X (gfx1250) — compile-verified
//
#include <hip/hip_runtime.h>
#include <math.h>

typedef __attribute__((ext_vector_type(2))) float v2f;
typedef __attribute__((ext_vector_type(8))) float v8f;

#define NEG_SLOPE 0.01f
#define DOUT 64
#define AGG_LOOKAHEAD 64   // edges ahead to prefetch z[src] rows

// ---------------------------------------------------------------------------
// Kernel 0: init out = 0, m = -inf, denom = 0
// ---------------------------------------------------------------------------
__global__ void gat_init_kernel(float* __restrict__ out, float* __restrict__ m,
                                float* __restrict__ denom, int n) {
    int i = blockIdx.x * blockDim.x + threadIdx.x;
    int total = n * DOUT;
    if (i < total) out[i] = 0.0f;
    if (i < n) {
        m[i] = -__builtin_inff();
        denom[i] = 0.0f;
    }
}

// ---------------------------------------------------------------------------
// Kernel 1: z = h @ W_fc^T + b_fc via V_WMMA_F32_16X16X4_F32
// One wave handles a 16-node row tile and all 64 outputs (4 N-tiles).
// A (16x4 f32): lanes 0-15 K=0,1 ; lanes 16-31 K=2,3 (float2 per lane)
// B (4x16 f32): same half-split over K; B[k][n] = W_fc[n][k]
// D (16x16 f32): VGPR v, lanes 0-15 -> M=v, lanes 16-31 -> M=v+8
// ---------------------------------------------------------------------------
__global__ void gat_proj_wmma_kernel(const float* __restrict__ h,
                                     const float* __restrict__ Wfc,
                                     const float* __restrict__ bfc,
                                     float* __restrict__ z, int n) {
    int wave = (blockIdx.x * blockDim.x + threadIdx.x) >> 5;
    int lane = threadIdx.x & 31;
    int tiles = (n + 15) >> 4;
    if (wave >= tiles) return;   // wave-uniform: EXEC stays all-1s for WMMA

    int row0 = wave << 4;
    int half = lane >> 4;        // 0: K low pair, 1: K high pair
    int l16  = lane & 15;
    int arow = row0 + l16;
    if (arow >= n) arow = n - 1; // clamp loads (only hit on a ragged last tile)
    bool full_tile = (row0 + 16) <= n;   // wave-uniform store predicate

    // Preload A fragments for all 16 k-steps (K = ks*4 + half*2 + {0,1})
    v2f afrag[16];
#pragma unroll
    for (int ks = 0; ks < 16; ++ks) {
        const float* ap = h + (size_t)arow * 64 + (ks << 2) + (half << 1);
        v2f a; a.x = ap[0]; a.y = ap[1];
        afrag[ks] = a;
    }

#pragma unroll
    for (int nt = 0; nt < 4; ++nt) {
        int col = (nt << 4) + l16;           // output feature index
        v8f acc = {};
#pragma unroll
        for (int ks = 0; ks < 16; ++ks) {
            const float* bp = Wfc + (size_t)col * 64 + (ks << 2) + (half << 1);
            v2f b; b.x = bp[0]; b.y = bp[1];
            acc = __builtin_amdgcn_wmma_f32_16x16x4_f32(
                false, afrag[ks], false, b, (short)0, acc, false, false);
        }
        float bias = bfc[col];
        if (full_tile) {
            // fast path: no per-element exec masking
#pragma unroll
            for (int v = 0; v < 8; ++v) {
                int row = row0 + v + (half << 3);
                z[(size_t)row * 64 + col] = acc[v] + bias;
            }
        } else {
#pragma unroll
            for (int v = 0; v < 8; ++v) {
                int row = row0 + v + (half << 3);
                if (row < n) z[(size_t)row * 64 + col] = acc[v] + bias;
            }
        }
    }
}

// ---------------------------------------------------------------------------
// Kernel 2: per-node attention scalars  es[i] = z[i]·a_src, ed[i] = z[i]·a_dst
// (factorized edge score: e = leaky(es[src] + ed[dst] + b))
// ---------------------------------------------------------------------------
__global__ void gat_node_dots_kernel(const float* __restrict__ z,
                                     const float* __restrict__ Wattn,
                                     float* __restrict__ es,
                                     float* __restrict__ ed, int n) {
    int i = blockIdx.x * blockDim.x + threadIdx.x;
    if (i >= n) return;
    const float4* zp = (const float4*)(z + (size_t)i * 64);
    const float4* ws = (const float4*)(Wattn);        // W_attn[0, 0:64]
    const float4* wd = (const float4*)(Wattn + 64);   // W_attn[0, 64:128]
    float s = 0.0f, d = 0.0f;
#pragma unroll
    for (int j = 0; j < 16; ++j) {
        float4 zv = zp[j];
        float4 a  = ws[j];
        float4 b  = wd[j];
        s += zv.x * a.x + zv.y * a.y + zv.z * a.z + zv.w * a.w;
        d += zv.x * b.x + zv.y * b.y + zv.z * b.z + zv.w * b.w;
    }
    es[i] = s;
    ed[i] = d;
}

// ---------------------------------------------------------------------------
// Sign-aware atomic float max (works with -inf init, mixes pos/neg correctly)
// ---------------------------------------------------------------------------
__device__ __forceinline__ void atomicMaxFloat(float* addr, float val) {
    if (val >= 0.0f)
        atomicMax((int*)addr, __float_as_int(val));
    else
        atomicMin((unsigned int*)addr, __float_as_uint(val));
}

// ---------------------------------------------------------------------------
// Kernel 3: edge scores + segment max over dst
// ---------------------------------------------------------------------------
__global__ void gat_edge_score_kernel(const float* __restrict__ es,
                                      const float* __restrict__ ed,
                                      const int* __restrict__ src,
                                      const int* __restrict__ dst,
                                      const float* __restrict__ battn,
                                      float* __restrict__ e_edge,
                                      float* __restrict__ m, int E) {
    int i = blockIdx.x * blockDim.x + threadIdx.x;
    if (i >= E) return;
    float e = es[src[i]] + ed[dst[i]] + battn[0];
    e = (e >= 0.0f) ? e : NEG_SLOPE * e;
    e_edge[i] = e;
    atomicMaxFloat(&m[dst[i]], e);
}

// ---------------------------------------------------------------------------
// Kernel 4: exp(e - m[dst]) in place + segment sum of denominators
// ---------------------------------------------------------------------------
__global__ void gat_edge_exp_kernel(const int* __restrict__ dst,
                                    const float* __restrict__ m,
                                    float* __restrict__ e_edge,
                                    float* __restrict__ denom, int E) {
    int i = blockIdx.x * blockDim.x + threadIdx.x;
    if (i >= E) return;
    int d = dst[i];
    float v = expf(e_edge[i] - m[d]);
    e_edge[i] = v;
    atomicAdd(&denom[d], v);
}

// ---------------------------------------------------------------------------
// Kernel 5: out[dst] += alpha * z[src]; 64 threads per edge (one per feature).
// Prefetch the z row of a lookahead edge to hide the random-gather latency
// (emits global_prefetch_b8; one byte per lane pulls the whole row's lines).
// ---------------------------------------------------------------------------
__global__ void gat_aggregate_kernel(const float* __restrict__ z,
                                     const float* __restrict__ e_edge,
                                     const float* __restrict__ denom,
                                     const int* __restrict__ src,
                                     const int* __restrict__ dst,
                                     float* __restrict__ out, int E) {
    int edge = blockIdx.x * blockDim.y + threadIdx.y;
    int t = threadIdx.x;                 // 0..63 feature lane
    if (edge >= E) return;

    int la = edge + AGG_LOOKAHEAD;
    if (la < E) {
        int s_next = src[la];
        __builtin_prefetch(&z[(size_t)s_next * 64 + t], 0, 1);
    }

    int s = src[edge];
    int d = dst[edge];
    float alpha = e_edge[edge] / denom[d];
    atomicAdd(&out[(size_t)d * 64 + t], alpha * z[(size_t)s * 64 + t]);
}

// ---------------------------------------------------------------------------
// Host launch
// ---------------------------------------------------------------------------
extern "C" void kernel_launch(void* const* d_in, const int* in_sizes, int n_in,
                              void* d_out, int out_size, void* d_ws, size_t ws_size,
                              hipStream_t stream) {
    const float* h     = (const float*)d_in[0];
    const float* Wfc   = (const float*)d_in[1];
    const float* bfc   = (const float*)d_in[2];
    const float* Wattn = (const float*)d_in[3];
    const float* battn = (const float*)d_in[4];
    const int*   src   = (const int*)d_in[5];
    const int*   dst   = (const int*)d_in[6];
    float* out = (float*)d_out;

    const int n = in_sizes[0] / DOUT;   // 50000
    const int E = in_sizes[5];          // 800000

    // Workspace layout (256B aligned slices)
    char* ws = (char*)d_ws;
    size_t off = 0;
    auto take = [&](size_t bytes) {
        void* p = ws + off;
        off = (off + bytes + 255) & ~(size_t)255;
        return p;
    };
    float* z      = (float*)take((size_t)n * DOUT * sizeof(float)); // 12.8 MB
    float* es     = (float*)take((size_t)n * sizeof(float));
    float* ed     = (float*)take((size_t)n * sizeof(float));
    float* m      = (float*)take((size_t)n * sizeof(float));
    float* denom  = (float*)take((size_t)n * sizeof(float));
    float* e_edge = (float*)take((size_t)E * sizeof(float));        // 3.2 MB

    // 0) init out / m / denom
    {
        int total = n * DOUT;
        gat_init_kernel<<<(total + 255) / 256, 256, 0, stream>>>(out, m, denom, n);
    }
    // 1) projection via WMMA: one wave per 16-node tile
    {
        int tiles = (n + 15) / 16;                 // 3125
        int waves_per_block = 4;                   // blockDim 128 = 4 waves
        int blocks = (tiles + waves_per_block - 1) / waves_per_block;
        gat_proj_wmma_kernel<<<blocks, waves_per_block * 32, 0, stream>>>(h, Wfc, bfc, z, n);
    }
    // 2) per-node attention scalars
    gat_node_dots_kernel<<<(n + 255) / 256, 256, 0, stream>>>(z, Wattn, es, ed, n);
    // 3) edge score + segment max
    gat_edge_score_kernel<<<(E + 255) / 256, 256, 0, stream>>>(es, ed, src, dst, battn,
                                                               e_edge, m, E);
    // 4) exp + segment sum
    gat_edge_exp_kernel<<<(E + 255) / 256, 256, 0, stream>>>(dst, m, e_edge, denom, E);
    // 5) weighted scatter-sum
    {
        dim3 block(64, 4);
        dim3 grid((E + 3) / 4);
        gat_aggregate_kernel<<<grid, block, 0, stream>>>(z, e_edge, denom, src, dst, out, E);
    }
}